// ResonantTunnelingAttention_68410239091114
// MI455X (gfx1250) — compile-verified
//
#include <hip/hip_runtime.h>
#include <hip/hip_bf16.h>
#include <math.h>

// ---------------- problem constants ----------------
#define BATCH   4
#define NWELLS  16
#define SEQ     2048
#define HID     512
#define TEMP_INV 10.0f
#define EPS_BW  1e-08f
#define EPS_NORM 1e-12f

// ---------------- vector types ----------------
typedef __attribute__((ext_vector_type(2))) float v2f;
typedef __attribute__((ext_vector_type(4))) float v4f;
typedef __attribute__((ext_vector_type(8))) float v8f;
typedef __attribute__((ext_vector_type(4))) unsigned int u32x4;
typedef __attribute__((ext_vector_type(8))) unsigned int u32x8;

// h-chunk per block and padded LDS stride: 20 = 16 data + 4 pad dwords.
// 20*s mod 64 distinct for s=0..15 -> conflict-free WMMA fragment gathers,
// and the TDM pad unit (pad_interval=16dw, pad_amount=4dw) reproduces it.
#define HC   16
#define PAD  20
#define TILE_F (16 * 16 * PAD)   // floats per LDS tile (5120)

// ---------------- kernel 1: zero partial sums ----------------
__global__ __launch_bounds__(256) void rt_zero_kernel(float* __restrict__ p, int n) {
    int i = blockIdx.x * blockDim.x + threadIdx.x;
    if (i < n) p[i] = 0.0f;
}

// ---------------- kernel 2: mean over (n, s) -> qsum[b*H+h] ----------------
__global__ __launch_bounds__(512) void rt_reduce_kernel(const float* __restrict__ W,
                                                        float* __restrict__ qsum) {
    const int h  = threadIdx.x;
    const int b  = blockIdx.y;
    const int s0 = blockIdx.x * 32;
    float acc = 0.0f;
    for (int n = 0; n < NWELLS; ++n) {
        const float* base = W + (((size_t)(b * NWELLS + n) * SEQ + s0) * HID) + h;
        #pragma unroll 4
        for (int s = 0; s < 32; ++s) {
            acc += base[(size_t)s * HID];
        }
    }
    atomicAdd(&qsum[b * HID + h], acc);
}

// ---------------- kernel 3: softmax over j of S_matrix/T ----------------
__global__ __launch_bounds__(512) void rt_softmax_kernel(const float* __restrict__ Smat,
                                                         float* __restrict__ ssoft) {
    const int h = threadIdx.x;
    const int i = blockIdx.x;
    float x[NWELLS];
    float m = -3.4e38f;
    #pragma unroll
    for (int j = 0; j < NWELLS; ++j) {
        x[j] = Smat[(i * NWELLS + j) * HID + h] * TEMP_INV;
        m = fmaxf(m, x[j]);
    }
    float sum = 0.0f;
    #pragma unroll
    for (int j = 0; j < NWELLS; ++j) {
        x[j] = __expf(x[j] - m);
        sum += x[j];
    }
    const float inv = 1.0f / sum;
    #pragma unroll
    for (int j = 0; j < NWELLS; ++j) {
        ssoft[(i * NWELLS + j) * HID + h] = x[j] * inv;
    }
}

// ---------------- kernel 4: tunneling amplitude ----------------
__global__ __launch_bounds__(512) void rt_amp_kernel(const float* __restrict__ qsum,
                                                     const float* __restrict__ E0,
                                                     const float* __restrict__ width,
                                                     float* __restrict__ amp_ws,
                                                     float* __restrict__ amp_out) {
    const int h = threadIdx.x;
    const int b = blockIdx.x;
    const float E = qsum[b * HID + h] * (1.0f / (float)(NWELLS * SEQ));
    float t[NWELLS];
    float l1 = 0.0f;
    #pragma unroll
    for (int n = 0; n < NWELLS; ++n) {
        const float e0 = E0[n * HID + h];
        const float g  = fabsf(width[n * HID + h]);
        const float d  = E - e0;
        const float den = d * d + (0.5f * g) * (0.5f * g) + EPS_BW;
        t[n] = g / den;
        l1 += fabsf(t[n]);
    }
    const float inv = 1.0f / fmaxf(l1, EPS_NORM);
    #pragma unroll
    for (int n = 0; n < NWELLS; ++n) {
        const float v = t[n] * inv;
        const int idx = (b * NWELLS + n) * HID + h;
        amp_ws[idx]  = v;
        amp_out[idx] = v;
    }
}

// ---------------- TDM: DMA one W tile [16j][16s][16h] -> padded LDS ----------
// 3D descriptor per CDNA5 ISA:
//   dim0 = h (contiguous, tile 16, tensor 512, stride 512)
//   dim1 = s (tile 16, tensor 2048, stride 512)
//   dim2 = j (tile 16, tensor 16, stride SEQ*HID)
// LDS pad: 4 dwords after every 16 dwords -> row stride PAD=20 floats.
__device__ __forceinline__ void tdm_load_w_tile(const float* gsrc, const float* ldst) {
    const unsigned long long ga = (unsigned long long)(uintptr_t)gsrc;
    const unsigned int la = (unsigned int)(uintptr_t)ldst;  // low 32b = LDS offset

    u32x4 g0;
    g0[0] = 0x1u;                                   // count=1, user descriptor
    g0[1] = la;                                     // lds_addr
    g0[2] = (unsigned int)(ga & 0xFFFFFFFFu);       // global_addr[31:0]
    g0[3] = (unsigned int)((ga >> 32) & 0x01FFFFFFu) | (2u << 30); // addr[56:32], type=2

    u32x8 g1;
    g1[0] = (2u << 16)        // data_size = 4B
          | (1u << 20)        // pad_enable
          | (3u << 22)        // pad_interval: 16 dwords
          | (3u << 25);       // pad_amount:   4 dwords
    g1[1] = ((unsigned)HID & 0xFFFFu) << 16;              // tensor_dim0 lo16
    g1[2] = ((unsigned)HID >> 16) | ((unsigned)SEQ << 16);// dim0 hi16 | dim1 lo16
    g1[3] = ((unsigned)SEQ >> 16) | (16u << 16);          // dim1 hi16 | tile_dim0
    g1[4] = 16u | (16u << 16);                            // tile_dim1 | tile_dim2
    g1[5] = (unsigned)HID;                                // dim0_stride[31:0]
    g1[6] = ((unsigned)((SEQ * (unsigned long long)HID) & 0xFFFFu)) << 16; // |dim1_stride lo16
    g1[7] = (unsigned)((SEQ * (unsigned long long)HID) >> 16);             // dim1_stride[47:16]

    u32x4 g2;
    g2[0] = 16u;                                    // tensor_dim2 (j)
    g2[1] = 0u;                                     // tensor_dim3 (unused)
    g2[2] = (unsigned)(SEQ * HID);                  // dim2_stride[31:0]
    g2[3] = 0u;                                     // dim2_stride hi | tile_dim3=0

    u32x4 g3 = {0u, 0u, 0u, 0u};

    asm volatile("tensor_load_to_lds %0, %1, %2, %3"
                 :: "s"(g0), "s"(g1), "s"(g2), "s"(g3)
                 : "memory");
}

// ---------------- kernel 5: WMMA mixing + amp scaling ----------------
// grid (H/HC = 32, SEQ/(16*8) = 16, B = 4), 256 threads (8 wave32)
// out[b,i,s,h] = amp[b,i,h] * sum_j ssoft[i,j,h] * W[b,j,s,h]
// W tiles arrive via TDM (double-buffered, wave 0 issues, TENSORcnt-tracked).
__global__ __launch_bounds__(256) void rt_mix_kernel(const float* __restrict__ W,
                                                     const float* __restrict__ ssoft,
                                                     const float* __restrict__ amp,
                                                     float* __restrict__ out) {
    extern __shared__ float smem[];
    float* lds_S   = smem;                 // [j][i][PAD]
    float* lds_Wb0 = smem + TILE_F;        // [j][s][PAD]  buffer 0
    float* lds_Wb1 = smem + 2 * TILE_F;    // [j][s][PAD]  buffer 1
    float* lds_O   = smem + 3 * TILE_F;    // [i][s][PAD]
    float* lds_amp = smem + 4 * TILE_F;    // [i][HC]

    const int h0   = blockIdx.x * HC;
    const int sblk = blockIdx.y;
    const int b    = blockIdx.z;
    const int t    = threadIdx.x;
    const int lane = t & 31;
    const int wave = t >> 5;

    const float* wbase = W + ((size_t)b * NWELLS * SEQ + (size_t)sblk * 8 * 16) * HID + h0;

    // kick off DMA of W tile 0 before doing our own staging loads
    if (wave == 0) {
        tdm_load_w_tile(wbase, lds_Wb0);
    }

    // load softmaxed S tile, transposed to [j][i][h] (coalesced b128)
    for (int q = t; q < 16 * 16 * (HC / 4); q += 256) {
        const int h4 = q % (HC / 4);
        const int ij = q / (HC / 4);
        const int j  = ij % 16;
        const int i  = ij / 16;
        const v4f v = *((const v4f*)(ssoft + (size_t)(i * NWELLS + j) * HID + h0) + h4);
        *(v4f*)&lds_S[(j * 16 + i) * PAD + 4 * h4] = v;
    }
    // amp tile (one float per thread)
    {
        const int h = t % HC;
        const int i = t / HC;
        lds_amp[i * HC + h] = amp[(size_t)(b * NWELLS + i) * HID + h0 + h];
    }
    if (wave == 0) __builtin_amdgcn_s_wait_tensorcnt(0);
    __syncthreads();

    const int nidx  = lane & 15;           // A: row M; B: col N
    const int kb    = (lane >> 4) << 1;    // K base 0 / 2
    const int mbase = (lane >> 4) << 3;    // C/D rows 0 / 8

    // ---- hoist loop-invariant operands into registers ----
    // A fragments (from lds_S) and amp scales depend only on (h, lane),
    // never on the s-tile; lds_S/lds_amp are read-only past this point.
    v2f   afrag[2][4];
    float ampreg[2][8];
    #pragma unroll
    for (int hh = 0; hh < 2; ++hh) {
        const int h = wave * 2 + hh;
        #pragma unroll
        for (int k = 0; k < 4; ++k) {
            const int j0 = 4 * k + kb;
            afrag[hh][k].x = lds_S[(j0       * 16 + nidx) * PAD + h];
            afrag[hh][k].y = lds_S[((j0 + 1) * 16 + nidx) * PAD + h];
        }
        #pragma unroll
        for (int r = 0; r < 8; ++r) {
            ampreg[hh][r] = lds_amp[(mbase + r) * HC + h];
        }
    }

    for (int st = 0; st < 8; ++st) {
        const float* ldsW = (st & 1) ? lds_Wb1 : lds_Wb0;
        // prefetch next W tile into the other buffer while we compute
        if (wave == 0 && st < 7) {
            tdm_load_w_tile(wbase + (size_t)(st + 1) * 16 * HID,
                            (st & 1) ? lds_Wb0 : lds_Wb1);
        }

        // per-h 16x16x16 product via 4 chained V_WMMA_F32_16X16X4_F32
        #pragma unroll
        for (int hh = 0; hh < 2; ++hh) {
            const int h = wave * 2 + hh;
            v8f c = {0.f, 0.f, 0.f, 0.f, 0.f, 0.f, 0.f, 0.f};
            #pragma unroll
            for (int k = 0; k < 4; ++k) {
                const int j0 = 4 * k + kb;
                v2f bb;
                bb.x = ldsW[(j0       * 16 + nidx) * PAD + h];
                bb.y = ldsW[((j0 + 1) * 16 + nidx) * PAD + h];
                c = __builtin_amdgcn_wmma_f32_16x16x4_f32(
                        false, afrag[hh][k], false, bb, (short)0, c, false, false);
            }
            #pragma unroll
            for (int r = 0; r < 8; ++r) {
                const int m = mbase + r;
                lds_O[(m * 16 + nidx) * PAD + h] = c[r] * ampreg[hh][r];
            }
        }
        __syncthreads();   // lds_O complete; all reads of ldsW done

        // streaming (nontemporal) b128 store of the output tile
        const int s0 = (sblk * 8 + st) * 16;
        for (int q = t; q < 16 * 16 * (HC / 4); q += 256) {
            const int h4 = q % (HC / 4);
            const int is = q / (HC / 4);
            const int s  = is % 16;
            const int i  = is / 16;
            const v4f v = *(const v4f*)&lds_O[(i * 16 + s) * PAD + 4 * h4];
            __builtin_nontemporal_store(
                v, (v4f*)(out + ((size_t)(b * NWELLS + i) * SEQ + s0 + s) * HID + h0) + h4);
        }
        if (wave == 0 && st < 7) __builtin_amdgcn_s_wait_tensorcnt(0);
        __syncthreads();   // next W tile visible; lds_O reusable
    }
}

// ---------------- launcher ----------------
extern "C" void kernel_launch(void* const* d_in, const int* in_sizes, int n_in,
                              void* d_out, int out_size, void* d_ws, size_t ws_size,
                              hipStream_t stream) {
    const float* W     = (const float*)d_in[0];   // (B, N, S, H)
    const float* Smat  = (const float*)d_in[1];   // (N, N, H)
    const float* E0    = (const float*)d_in[2];   // (N, H)
    const float* width = (const float*)d_in[3];   // (N, H)

    float* out     = (float*)d_out;                                   // (B,N,S,H)
    float* out_amp = out + (size_t)BATCH * NWELLS * SEQ * HID;        // (B,N,H)

    float* ws    = (float*)d_ws;
    float* qsum  = ws;                                  // B*H        = 2048
    float* amp   = ws + BATCH * HID;                    // B*N*H      = 32768
    float* ssoft = amp + BATCH * NWELLS * HID;          // N*N*H      = 131072

    rt_zero_kernel<<<dim3((BATCH * HID + 255) / 256), 256, 0, stream>>>(qsum, BATCH * HID);
    rt_reduce_kernel<<<dim3(SEQ / 32, BATCH), 512, 0, stream>>>(W, qsum);
    rt_softmax_kernel<<<dim3(NWELLS), 512, 0, stream>>>(Smat, ssoft);
    rt_amp_kernel<<<dim3(BATCH), 512, 0, stream>>>(qsum, E0, width, amp, out_amp);

    const size_t mix_smem = (size_t)(4 * TILE_F + NWELLS * HC) * sizeof(float); // ~81 KB
    rt_mix_kernel<<<dim3(HID / HC, SEQ / (16 * 8), BATCH), 256, mix_smem, stream>>>(
        W, ssoft, amp, out);
}